// StandardGeometry_28802050687696
// MI455X (gfx1250) — compile-verified
//
#include <hip/hip_runtime.h>
#include <math.h>

// ---------------------------------------------------------------------------
// Fan-beam CT gradient step: out = x - A^T (A x - p)
//   A  : ray-driven forward projector, 2 x 360 x 448 rays, 452 samples/ray,
//        bilinear interpolation over a 320x320 image.
// Fused design: a ray's residual depends only on its own forward sum, so
// forward march + residual + adjoint scatter fuse into one ray kernel.
// Wave32 layout: 16 rays per wave, 4 sample-phase slots per ray == the
// documented 16x4 f32 A-matrix layout, reduced with V_WMMA_F32_16X16X4_F32
// against an all-ones B matrix (layout-independent reduction).
// ---------------------------------------------------------------------------

namespace {
constexpr int    kN        = 320;
constexpr int    kAngles   = 360;
constexpr int    kDet      = 448;                       // 28 * 16
constexpr int    kSamp     = 452;                       // divisible by 4
constexpr int    kBatch    = 2;
constexpr float  kSID      = 500.0f;
constexpr float  kSDD      = 1000.0f;
constexpr float  kDU       = 1.6f;
constexpr double kR_d      = 226.27416997969521951;     // 160*sqrt(2)
constexpr float  kStep     = (float)(2.0 * kR_d / (double)kSamp);  // == DT
constexpr float  kDT       = kStep;
constexpr float  kT0       = (float)(500.0 - kR_d);
constexpr float  kHalf     = 159.5f;                    // (N-1)/2
constexpr float  kDetCtr   = 223.5f;                    // (DET-1)/2
constexpr float  kDeg2Rad  = (float)(3.14159265358979323846 / 180.0);
constexpr int    kDetGrps  = kDet / 16;                 // 28
constexpr int    kWavesPB  = 8;                         // 256-thread blocks
}

typedef __attribute__((ext_vector_type(2))) float v2f;
typedef __attribute__((ext_vector_type(8))) float v8f;

// ds_swizzle SWAPX16: group-of-32 mode, xor_mask=0x10, and_mask=0x1f.
__device__ __forceinline__ float swap_x16(float v) {
    return __int_as_float(__builtin_amdgcn_ds_swizzle(__float_as_int(v), 0x401F));
}

// Branchless bilinear gather: clamp indices (always-valid loads), mask weights.
__device__ __forceinline__ float bilin_gather(const float* __restrict__ v,
                                              float px, float py) {
    float x0 = floorf(px), y0 = floorf(py);
    float fx = px - x0,    fy = py - y0;
    int   xi = (int)x0,    yi = (int)y0;
    float gx = 1.0f - fx,  gy = 1.0f - fy;

    int x0c = min(max(xi,     0), kN - 1);
    int x1c = min(max(xi + 1, 0), kN - 1);
    int y0c = min(max(yi,     0), kN - 1);
    int y1c = min(max(yi + 1, 0), kN - 1);

    float v00 = v[y0c * kN + x0c];
    float v10 = v[y0c * kN + x1c];
    float v01 = v[y1c * kN + x0c];
    float v11 = v[y1c * kN + x1c];

    bool bx0 = (unsigned)xi       < (unsigned)kN;
    bool bx1 = (unsigned)(xi + 1) < (unsigned)kN;
    bool by0 = (unsigned)yi       < (unsigned)kN;
    bool by1 = (unsigned)(yi + 1) < (unsigned)kN;

    float w00 = (bx0 && by0) ? gx * gy : 0.0f;
    float w10 = (bx1 && by0) ? fx * gy : 0.0f;
    float w01 = (bx0 && by1) ? gx * fy : 0.0f;
    float w11 = (bx1 && by1) ? fx * fy : 0.0f;

    return fmaf(w00, v00, fmaf(w10, v10, fmaf(w01, v01, w11 * v11)));
}

// Adjoint scatter: one containment check; per-corner masking on the *value*
// (atomicAdd of +0.0f to a clamped address is a numeric no-op).
__device__ __forceinline__ void bilin_scatter(float* __restrict__ bp,
                                              float px, float py, float c) {
    float x0 = floorf(px), y0 = floorf(py);
    float fx = px - x0,    fy = py - y0;
    int   xi = (int)x0,    yi = (int)y0;
    if (xi < -1 || xi >= kN || yi < -1 || yi >= kN) return;
    float gx = 1.0f - fx,  gy = 1.0f - fy;

    int x0c = max(xi, 0),          y0c = max(yi, 0);
    int x1c = min(xi + 1, kN - 1), y1c = min(yi + 1, kN - 1);
    bool bx0 = xi >= 0,      by0 = yi >= 0;
    bool bx1 = xi < kN - 1,  by1 = yi < kN - 1;

    unsafeAtomicAdd(&bp[y0c * kN + x0c], (bx0 && by0) ? c * gx * gy : 0.0f);
    unsafeAtomicAdd(&bp[y0c * kN + x1c], (bx1 && by0) ? c * fx * gy : 0.0f);
    unsafeAtomicAdd(&bp[y1c * kN + x0c], (bx0 && by1) ? c * gx * fy : 0.0f);
    unsafeAtomicAdd(&bp[y1c * kN + x1c], (bx1 && by1) ? c * fx * fy : 0.0f);
}

// One wave handles 16 rays (same angle, 16 consecutive detectors).
// Lane L: ray m = L%16, phase H = L/16; accumulates sample classes
// s%4 == 2H (->a0) and s%4 == 2H+1 (->a1). (a0,a1) is exactly the 16x4 f32
// WMMA A-matrix layout; D = A x ones + 0 gives the full per-ray sum.
__global__ __launch_bounds__(256) void ct_ray_kernel(
    const float* __restrict__ vol,   // [2,320,320]
    const float* __restrict__ sino,  // [2,360,448]  (measured data p)
    float* __restrict__ bp)          // [2,320,320]  (pre-zeroed)
{
    const int lane   = threadIdx.x & 31;
    const int waveIn = threadIdx.x >> 5;
    const int w      = blockIdx.x * kWavesPB + waveIn;   // 0 .. 20159 exactly

    const int b   = w / (kAngles * kDetGrps);
    const int rem = w % (kAngles * kDetGrps);
    const int ang = rem / kDetGrps;
    const int dg  = rem % kDetGrps;

    const int m   = lane & 15;      // ray within wave
    const int H   = lane >> 4;      // K-slot half
    const int det = dg * 16 + m;

    // ---- per-ray geometry ------------------------------------------------
    const float beta = (float)ang * kDeg2Rad;
    const float cb = cosf(beta), sb = sinf(beta);
    const float u  = ((float)det - kDetCtr) * kDU;
    float dx = -kSDD * cb - u * sb;
    float dy = -kSDD * sb + u * cb;
    const float inv = 1.0f / sqrtf(kSDD * kSDD + u * u);
    dx *= inv; dy *= inv;
    const float ox = kSID * cb + kHalf;   // src + half, so px = ox + t*dx
    const float oy = kSID * sb + kHalf;

    const float* v = vol + b * kN * kN;
    const float sBase = (float)(2 * H) + 0.5f;  // first sample class + 0.5

    // ---- forward march: two partial sums per lane ------------------------
    float a0 = 0.0f, a1 = 0.0f;
    {
        float sf = sBase;                        // exact integer+0.5 counter
        #pragma unroll 2
        for (int i = 0; i < kSamp / 4; ++i) {
            float t0 = fmaf(sf,        kStep, kT0);
            float t1 = fmaf(sf + 1.0f, kStep, kT0);
            a0 += bilin_gather(v, fmaf(t0, dx, ox), fmaf(t0, dy, oy));
            a1 += bilin_gather(v, fmaf(t1, dx, ox), fmaf(t1, dy, oy));
            sf += 4.0f;                          // exact
        }
    }

    // ---- WMMA reduction: D[r,n] = sum_k A[r,k]  (B = ones, C = 0) --------
    v2f A; A.x = a0; A.y = a1;
    v2f B; B.x = 1.0f; B.y = 1.0f;
    v8f C = {};
    C = __builtin_amdgcn_wmma_f32_16x16x4_f32(
            /*neg_a=*/false, A, /*neg_b=*/false, B,
            /*c_mod=*/(short)0, C, /*reuse_a=*/false, /*reuse_b=*/false);

    // D layout: VGPR j, lanes 0-15 -> D[j, lane]; lanes 16-31 -> D[8+j, lane-16].
    // Select element (m&7) with a cndmask tree (no dynamic index -> no LDS
    // spill), then swap halves with ds_swizzle SWAPX16 when row-half != H.
    float mine;
    {
        const int k = m & 7;
        const bool b0 = (k & 1) != 0;
        const bool b1 = (k & 2) != 0;
        const bool b2 = (k & 4) != 0;
        float x01 = b0 ? C[1] : C[0];
        float x23 = b0 ? C[3] : C[2];
        float x45 = b0 ? C[5] : C[4];
        float x67 = b0 ? C[7] : C[6];
        float x03 = b1 ? x23 : x01;
        float x47 = b1 ? x67 : x45;
        mine = b2 ? x47 : x03;
    }
    float other  = swap_x16(mine);
    float raysum = (((lane >> 3) & 1) == H) ? mine : other;

    // ---- residual for this lane's ray ------------------------------------
    const float meas = sino[(b * kAngles + ang) * kDet + det];
    const float res  = fmaf(raysum, kDT, -meas);   // A x - p
    const float coef = res * kDT;                  // adjoint carries DT too

    // ---- adjoint march: scatter with identical weights -------------------
    float* bpv = bp + b * kN * kN;
    {
        float sf = sBase;
        #pragma unroll 2
        for (int i = 0; i < kSamp / 4; ++i) {
            float t0 = fmaf(sf,        kStep, kT0);
            float t1 = fmaf(sf + 1.0f, kStep, kT0);
            bilin_scatter(bpv, fmaf(t0, dx, ox), fmaf(t0, dy, oy), coef);
            bilin_scatter(bpv, fmaf(t1, dx, ox), fmaf(t1, dy, oy), coef);
            sf += 4.0f;
        }
    }
}

__global__ void ct_zero_kernel(float* __restrict__ bp, int n) {
    int i = blockIdx.x * blockDim.x + threadIdx.x;
    if (i < n) bp[i] = 0.0f;
}

__global__ void ct_axpy_kernel(const float* __restrict__ x,
                               const float* __restrict__ bp,
                               float* __restrict__ out, int n) {
    int i = blockIdx.x * blockDim.x + threadIdx.x;
    if (i < n) out[i] = x[i] - bp[i];   // LAMB = 1.0
}

extern "C" void kernel_launch(void* const* d_in, const int* in_sizes, int n_in,
                              void* d_out, int out_size, void* d_ws, size_t ws_size,
                              hipStream_t stream) {
    const float* x = (const float*)d_in[0];   // [2,320,320] f32
    const float* p = (const float*)d_in[1];   // [2,360,448] f32
    float* out = (float*)d_out;               // [2,320,320] f32
    float* bp  = (float*)d_ws;                // backprojection accumulator

    const int nvox = kBatch * kN * kN;        // 204800

    ct_zero_kernel<<<(nvox + 255) / 256, 256, 0, stream>>>(bp, nvox);

    const int totalWaves = kBatch * kAngles * kDetGrps;   // 20160
    ct_ray_kernel<<<totalWaves / kWavesPB, 256, 0, stream>>>(x, p, bp);

    ct_axpy_kernel<<<(nvox + 255) / 256, 256, 0, stream>>>(x, bp, out, nvox);
}